// GAT_JK_11424613007588
// MI455X (gfx1250) — compile-verified
//
#include <hip/hip_runtime.h>
#include <hip/hip_bf16.h>

// Problem constants (fixed by the reference).
#define N_NODES 50000
#define E_EDGES 800000
#define EP      (E_EDGES + N_NODES)   // edges + self loops = 850000
#define IN_CH   128
#define HID     256
#define HEADS   8
#define CPH     32                    // per-head channels
#define NCLS    64
#define OUTW    (4 * HID + NCLS)      // 1088 JK-concat width

typedef float v2f __attribute__((ext_vector_type(2)));
typedef float v4f __attribute__((ext_vector_type(4)));
typedef float v8f __attribute__((ext_vector_type(8)));

// ---------------------------------------------------------------------------
// C[M x Ncols] = A[M x K] @ B[K x Ncols] (+ bias per column), f32 WMMA.
// One wave per 16x16 output tile; V_WMMA_F32_16X16X4_F32 chain over K.
// ---------------------------------------------------------------------------
__global__ __launch_bounds__(256)
void gemm_wmma_f32(const float* __restrict__ A, int lda,
                   const float* __restrict__ B, int ldb,
                   float* __restrict__ C, int ldc,
                   int M, int K, int Ncols,
                   const float* __restrict__ bias)
{
    const int wave = threadIdx.x >> 5;
    const int lane = threadIdx.x & 31;
    const int ntc  = Ncols >> 4;
    const int T    = (M >> 4) * ntc;
    const int tile = blockIdx.x * 8 + wave;
    if (tile >= T) return;                 // wave-uniform: EXEC stays all-ones

    const int tm = tile / ntc, tn = tile - tm * ntc;
    const int row0 = tm << 4, col0 = tn << 4;
    const int m    = lane & 15;            // row within tile (A) / col (B,C)
    const int half = lane >> 4;

    v8f acc;
    const float binit = bias ? bias[col0 + m] : 0.0f;  // bias depends on col only
#pragma unroll
    for (int i = 0; i < 8; ++i) acc[i] = binit;

    const float* Arow = A + (size_t)(row0 + m) * lda + 2 * half;
    const float* Bcol = B + (size_t)(2 * half) * ldb + col0 + m;

#pragma unroll 4
    for (int k0 = 0; k0 < K; k0 += 4) {
        v2f a = *(const v2f*)(Arow + k0);
        v2f b;
        b.x = Bcol[(size_t)k0 * ldb];
        b.y = Bcol[(size_t)(k0 + 1) * ldb];
        acc = __builtin_amdgcn_wmma_f32_16x16x4_f32(
                  false, a, false, b, (short)0, acc, false, false);
    }

    float* Crow = C + col0 + m;
#pragma unroll
    for (int i = 0; i < 8; ++i)
        Crow[(size_t)(row0 + i + 8 * half) * ldc] = acc[i];
}

// ---------------------------------------------------------------------------
// CSR build: degree count -> exclusive scan (LDS) -> placement.
// deg/rowptr share one buffer. Edge ids >= E are the implicit self loops.
// ---------------------------------------------------------------------------
__global__ __launch_bounds__(256)
void k_zero_i32(int* __restrict__ p, int n)
{
    int i = blockIdx.x * blockDim.x + threadIdx.x;
    if (i < n) p[i] = 0;
}

__global__ __launch_bounds__(256)
void k_count(const long long* __restrict__ ei, int* __restrict__ deg)
{
    int e = blockIdx.x * blockDim.x + threadIdx.x;
    if (e >= EP) return;
    int d = (e < E_EDGES) ? (int)ei[E_EDGES + e] : (e - E_EDGES);
    atomicAdd(&deg[d], 1);
}

__global__ __launch_bounds__(256)
void k_scan1(int* __restrict__ data, int* __restrict__ aux, int n)
{
    __shared__ int sm[256];
    const int tid = threadIdx.x;
    const int i   = blockIdx.x * 256 + tid;
    const int v   = (i < n) ? data[i] : 0;
    sm[tid] = v;
    __syncthreads();
#pragma unroll
    for (int off = 1; off < 256; off <<= 1) {
        int t = (tid >= off) ? sm[tid - off] : 0;
        __syncthreads();
        sm[tid] += t;
        __syncthreads();
    }
    if (i < n) data[i] = sm[tid] - v;          // exclusive within block
    if (tid == 255) aux[blockIdx.x] = sm[tid]; // block total
}

__global__ __launch_bounds__(256)
void k_scan2(int* __restrict__ aux, int nb)    // single block, nb <= 256
{
    __shared__ int sm[256];
    const int tid = threadIdx.x;
    const int v   = (tid < nb) ? aux[tid] : 0;
    sm[tid] = v;
    __syncthreads();
#pragma unroll
    for (int off = 1; off < 256; off <<= 1) {
        int t = (tid >= off) ? sm[tid - off] : 0;
        __syncthreads();
        sm[tid] += t;
        __syncthreads();
    }
    aux[tid] = sm[tid] - v;                    // exclusive block offsets
}

__global__ __launch_bounds__(256)
void k_scan3(int* __restrict__ rowptr, const int* __restrict__ aux,
             int* __restrict__ cursor, int n)
{
    int i = blockIdx.x * blockDim.x + threadIdx.x;
    if (i < n) {
        int r = rowptr[i] + aux[i >> 8];
        rowptr[i] = r;
        cursor[i] = r;
    }
    if (i == 0) rowptr[n] = EP;                // total edge count is fixed
}

__global__ __launch_bounds__(256)
void k_place(const long long* __restrict__ ei, int* __restrict__ cursor,
             int* __restrict__ csr_src)
{
    int e = blockIdx.x * blockDim.x + threadIdx.x;
    if (e >= EP) return;
    int s, d;
    if (e < E_EDGES) { s = (int)ei[e]; d = (int)ei[E_EDGES + e]; }
    else             { s = e - E_EDGES; d = s; }
    int pos = atomicAdd(&cursor[d], 1);
    csr_src[pos] = s;
}

// ---------------------------------------------------------------------------
// Per (node, head): alpha_src / alpha_dst dot products.
// ---------------------------------------------------------------------------
__global__ __launch_bounds__(256)
void alpha_k(const float* __restrict__ hw,
             const float* __restrict__ a_src, const float* __restrict__ a_dst,
             float* __restrict__ as, float* __restrict__ ad,
             int heads, int C, int n)
{
    const int idx = blockIdx.x * blockDim.x + threadIdx.x;
    if (idx >= n * heads) return;
    const int node = idx / heads, h = idx - node * heads;
    const float* hp  = hw + (size_t)node * heads * C + h * C;
    const float* asp = a_src + h * C;
    const float* adp = a_dst + h * C;
    float s = 0.f, d = 0.f;
    for (int c = 0; c < C; ++c) { float v = hp[c]; s += v * asp[c]; d += v * adp[c]; }
    as[idx] = s;
    ad[idx] = d;
}

// ---------------------------------------------------------------------------
// One wave per destination node; zero-atomic CSR gather aggregation.
// Lane l owns channels [l*VPT, l*VPT+VPT); its head is (l*VPT)/C (uniform
// because VPT divides C). Pass 1 = per-head segment max (lane-strided edges +
// shfl_xor reduce, matching the reference softmax numerics). Pass 2 = uniform
// edge loop: one exp per lane, 128-bit gathers of hW[src] (L2-resident:
// hW is 51 MB vs 192 MB L2), register accumulation, fused normalize+bias+ReLU.
// ---------------------------------------------------------------------------
template<int H, int C>
__global__ __launch_bounds__(256)
void gat_aggregate(const int* __restrict__ rowptr, const int* __restrict__ csr_src,
                   const float* __restrict__ as, const float* __restrict__ ad,
                   const float* __restrict__ hw,        // [N, H*C]
                   float* __restrict__ outp, int ldo,   // dst JK slice, stride OUTW
                   const float* __restrict__ bias, int relu, int n)
{
    constexpr int HC  = H * C;
    constexpr int VPT = HC / 32;           // floats per lane (8 or 2)
    const int wave = threadIdx.x >> 5;
    const int lane = threadIdx.x & 31;
    const int d = blockIdx.x * 8 + wave;
    if (d >= n) return;                    // wave-uniform

    const int r0 = rowptr[d];
    const int r1 = rowptr[d + 1];
    const int head = (lane * VPT) / C;     // fixed head per lane

    float ad_h[H];
    if constexpr (H == 8) {
        v4f a0 = *(const v4f*)(ad + (size_t)d * 8);
        v4f a1 = *(const v4f*)(ad + (size_t)d * 8 + 4);
        ad_h[0]=a0.x; ad_h[1]=a0.y; ad_h[2]=a0.z; ad_h[3]=a0.w;
        ad_h[4]=a1.x; ad_h[5]=a1.y; ad_h[6]=a1.z; ad_h[7]=a1.w;
    } else {
        ad_h[0] = ad[d];
    }

    // ---- Pass 1: segment max per head ----
    float mx[H];
#pragma unroll
    for (int h = 0; h < H; ++h) mx[h] = -INFINITY;
    for (int i = r0 + lane; i < r1; i += 32) {
        const int s = csr_src[i];
        float av[H];
        if constexpr (H == 8) {
            v4f a0 = *(const v4f*)(as + (size_t)s * 8);
            v4f a1 = *(const v4f*)(as + (size_t)s * 8 + 4);
            av[0]=a0.x; av[1]=a0.y; av[2]=a0.z; av[3]=a0.w;
            av[4]=a1.x; av[5]=a1.y; av[6]=a1.z; av[7]=a1.w;
        } else {
            av[0] = as[s];
        }
#pragma unroll
        for (int h = 0; h < H; ++h) {
            float x = av[h] + ad_h[h];
            float e = x > 0.f ? x : 0.2f * x;
            mx[h] = fmaxf(mx[h], e);
        }
    }
#pragma unroll
    for (int h = 0; h < H; ++h)
#pragma unroll
        for (int off = 16; off; off >>= 1)
            mx[h] = fmaxf(mx[h], __shfl_xor(mx[h], off, 32));

    float mxl = mx[0], adl = ad_h[0];
#pragma unroll
    for (int h = 1; h < H; ++h)
        if (head == h) { mxl = mx[h]; adl = ad_h[h]; }

    // ---- Pass 2: gather + accumulate (no atomics) ----
    float acc[VPT] = {};
    float den = 0.f;
    for (int i = r0; i < r1; ++i) {
        int s = csr_src[i];
        s = __builtin_amdgcn_readfirstlane(s);     // uniform: all lanes read same i
        float x = as[(size_t)s * H + head] + adl;
        float e = x > 0.f ? x : 0.2f * x;
        float w = __expf(e - mxl);
        den += w;
        const float* hp = hw + (size_t)s * HC + lane * VPT;
        if constexpr (VPT == 8) {
            v4f h0 = *(const v4f*)hp;
            v4f h1 = *(const v4f*)(hp + 4);
            acc[0] += w * h0.x; acc[1] += w * h0.y; acc[2] += w * h0.z; acc[3] += w * h0.w;
            acc[4] += w * h1.x; acc[5] += w * h1.y; acc[6] += w * h1.z; acc[7] += w * h1.w;
        } else {
            v2f h0 = *(const v2f*)hp;
            acc[0] += w * h0.x; acc[1] += w * h0.y;
        }
    }

    const float inv = 1.f / (den + 1e-16f);   // identical across lanes of a head
    float* op = outp + (size_t)d * ldo + lane * VPT;
#pragma unroll
    for (int t = 0; t < VPT; ++t) {
        float v = acc[t] * inv + bias[lane * VPT + t];
        if (relu) v = fmaxf(v, 0.f);
        op[t] = v;
    }
}

// ---------------------------------------------------------------------------
extern "C" void kernel_launch(void* const* d_in, const int* in_sizes, int n_in,
                              void* d_out, int out_size, void* d_ws, size_t ws_size,
                              hipStream_t stream)
{
    const float*     x   = (const float*)d_in[0];
    const long long* ei  = (const long long*)d_in[1];   // edge_index is int64
    const float*     W0  = (const float*)d_in[2];
    const float*     b0  = (const float*)d_in[3];
    const float*     Wc  = (const float*)d_in[4];
    const float*     asc = (const float*)d_in[5];
    const float*     adc = (const float*)d_in[6];
    const float*     bc  = (const float*)d_in[7];
    const float*     Wl  = (const float*)d_in[8];
    const float*     asl = (const float*)d_in[9];
    const float*     adl = (const float*)d_in[10];
    const float*     bl  = (const float*)d_in[11];
    float* out = (float*)d_out;                          // [N, 1088]

    // Workspace (~58 MB): hw | as | ad | rowptr | cursor | aux | csr_src
    char* ws = (char*)d_ws;
    float* hw     = (float*)ws;                                         // N*HID
    float* asb    = (float*)(ws + (size_t)N_NODES * HID * 4);           // N*HEADS
    float* adb    = asb + (size_t)N_NODES * HEADS;                      // N*HEADS
    int*   rowptr = (int*)(adb + (size_t)N_NODES * HEADS);              // N+1
    int*   cursor = rowptr + (N_NODES + 1);                             // N
    int*   aux    = cursor + N_NODES;                                   // 256
    int*   csr    = aux + 256;                                          // EP

    const dim3 blk(256);
    const int nEdgeBlocks = (EP + 255) / 256;
    const int nScanBlocks = (N_NODES + 255) / 256;   // 196 <= 256

    // ---- Build dst-sorted CSR once (graph is shared by all 4 layers) ----
    k_zero_i32<<<(N_NODES + 1 + 255) / 256, blk, 0, stream>>>(rowptr, N_NODES + 1);
    k_count<<<nEdgeBlocks, blk, 0, stream>>>(ei, rowptr);
    k_scan1<<<nScanBlocks, blk, 0, stream>>>(rowptr, aux, N_NODES);
    k_scan2<<<1, blk, 0, stream>>>(aux, nScanBlocks);
    k_scan3<<<nScanBlocks, blk, 0, stream>>>(rowptr, aux, cursor, N_NODES);
    k_place<<<nEdgeBlocks, blk, 0, stream>>>(ei, cursor, csr);

    // h0 = x @ W0 + b0 -> JK slice [0, 256)  (also the layer-1 input)
    {
        int T = (N_NODES / 16) * (HID / 16);
        gemm_wmma_f32<<<(T + 7) / 8, blk, 0, stream>>>(
            x, IN_CH, W0, HID, out, OUTW, N_NODES, IN_CH, HID, b0);
    }

    // 3 hidden GATConv layers (8 heads x 32 ch, concat, ReLU)
    for (int l = 0; l < 3; ++l) {
        const float* hin = out + (size_t)l * HID;        // col slice, stride OUTW
        float* hnext     = out + (size_t)(l + 1) * HID;

        int T = (N_NODES / 16) * (HID / 16);
        gemm_wmma_f32<<<(T + 7) / 8, blk, 0, stream>>>(
            hin, OUTW, Wc + (size_t)l * HID * HID, HID, hw, HID,
            N_NODES, HID, HID, nullptr);

        int na = N_NODES * HEADS;
        alpha_k<<<(na + 255) / 256, blk, 0, stream>>>(
            hw, asc + l * HEADS * CPH, adc + l * HEADS * CPH,
            asb, adb, HEADS, CPH, N_NODES);

        gat_aggregate<HEADS, CPH><<<(N_NODES + 7) / 8, blk, 0, stream>>>(
            rowptr, csr, asb, adb, hw, hnext, OUTW,
            bc + (size_t)l * HID, /*relu=*/1, N_NODES);
    }

    // Final GATConv: 1 head, 64 classes, no relu -> JK slice [1024, 1088)
    {
        const float* hin = out + (size_t)3 * HID;
        float* hout      = out + (size_t)4 * HID;

        int T = (N_NODES / 16) * (NCLS / 16);
        gemm_wmma_f32<<<(T + 7) / 8, blk, 0, stream>>>(
            hin, OUTW, Wl, NCLS, hw, NCLS, N_NODES, HID, NCLS, nullptr);

        alpha_k<<<(N_NODES + 255) / 256, blk, 0, stream>>>(
            hw, asl, adl, asb, adb, 1, NCLS, N_NODES);

        gat_aggregate<1, NCLS><<<(N_NODES + 7) / 8, blk, 0, stream>>>(
            rowptr, csr, asb, adb, hw, hout, OUTW, bl, /*relu=*/0, N_NODES);
    }
}